// GCN_11098195493276
// MI455X (gfx1250) — compile-verified
//
#include <hip/hip_runtime.h>
#include <hip/hip_bf16.h>

// ---------------------------------------------------------------------------
// 2-layer GCN for MI455X (gfx1250, wave32).
//  - Dense transform x@W1 uses V_WMMA_F32_16X16X4_F32 (full fp32 precision).
//  - Edge aggregation uses fire-and-forget global_atomic_add_f32 (inline asm,
//    non-returning -> STOREcnt-tracked L2 RMW, no CAS loop possible).
//  - Whole working set (~70 MB) fits in the 192 MB L2.
// ---------------------------------------------------------------------------

typedef __attribute__((ext_vector_type(2))) float v2f;
typedef __attribute__((ext_vector_type(8))) float v8f;

#define F1 16   // hidden features
#define F2 2    // output features
#define FIN 25  // input features

__device__ __forceinline__ void atomic_add_f32(float* p, float v) {
    // Non-returning fp32 atomic add: guaranteed native L2 RMW on gfx1250.
    // (TH[0]=0 -> no pre-op value returned; tracked by STOREcnt; S_ENDPGM's
    //  implicit wait-idle guarantees completion before kernel end.)
    asm volatile("global_atomic_add_f32 %0, %1, off"
                 :
                 : "v"(p), "v"(v)
                 : "memory");
}

// ---- init: deg = 1.0 (self loop), accumulators = 0 -------------------------
__global__ void gcn_init_kernel(float* __restrict__ deg,
                                float* __restrict__ agg1,
                                float* __restrict__ agg2,
                                int n) {
    int gid = blockIdx.x * blockDim.x + threadIdx.x;
    if (gid < 16 * n) agg1[gid] = 0.0f;
    if (gid < 2 * n)  agg2[gid] = 0.0f;
    if (gid < n)      deg[gid]  = 1.0f;   // self-loop contributes 1 to degree
}

// ---- degree: deg[col[e]] += 1 ---------------------------------------------
__global__ void gcn_degree_kernel(const int* __restrict__ cols,
                                  float* __restrict__ deg,
                                  int n_edges) {
    int e = blockIdx.x * blockDim.x + threadIdx.x;
    if (e >= n_edges) return;
    atomic_add_f32(&deg[cols[e]], 1.0f);
}

// ---- dinv = rsqrt(deg) (deg >= 1 always) ----------------------------------
__global__ void gcn_dinv_kernel(const float* __restrict__ deg,
                                float* __restrict__ dinv,
                                int n) {
    int i = blockIdx.x * blockDim.x + threadIdx.x;
    if (i >= n) return;
    dinv[i] = rsqrtf(deg[i]);
}

// ---- h1 = x @ W1 via V_WMMA_F32_16X16X4_F32 -------------------------------
// One wave handles a 16-node x 16-feature tile; 7 K-steps of 4 cover K=25
// (zero-padded). Fragment layouts per CDNA5 ISA 7.12.2:
//   A 16x4 f32:  lane m=(L&15); half=(L>>4): VGPR0=K(2*half), VGPR1=K(2*half+1)
//   B  4x16 f32: lane n=(L&15); half selects K pair (mirror of A)
//   C/D 16x16:   VGPR v holds M = v + 8*half, N = (L&15)
__global__ void gcn_xw1_wmma_kernel(const float* __restrict__ x,   // [N,25]
                                    const float* __restrict__ W1,  // [25,16]
                                    float* __restrict__ h1,        // [N,16]
                                    int n_nodes) {
    const int lane  = threadIdx.x & 31;
    const int wave  = threadIdx.x >> 5;
    const int tile  = blockIdx.x * (blockDim.x >> 5) + wave;
    const int node0 = tile * 16;                    // N_NODES % 16 == 0
    if (node0 >= n_nodes) return;                   // whole wave exits together

    const int half = lane >> 4;
    const int l16  = lane & 15;
    const int m    = l16;

    v8f c = {0.f, 0.f, 0.f, 0.f, 0.f, 0.f, 0.f, 0.f};

    #pragma unroll
    for (int k0 = 0; k0 < 28; k0 += 4) {            // k0 = 0,4,...,24
        const int ka = k0 + 2 * half;
        v2f a, b;
        a.x = (ka     < FIN) ? x[(node0 + m) * FIN + ka    ] : 0.0f;
        a.y = (ka + 1 < FIN) ? x[(node0 + m) * FIN + ka + 1] : 0.0f;
        b.x = (ka     < FIN) ? W1[(ka    ) * F1 + l16] : 0.0f;
        b.y = (ka + 1 < FIN) ? W1[(ka + 1) * F1 + l16] : 0.0f;
        c = __builtin_amdgcn_wmma_f32_16x16x4_f32(
                /*neg_a=*/false, a, /*neg_b=*/false, b,
                /*c_mod=*/(short)0, c, /*reuse_a=*/false, /*reuse_b=*/false);
    }

    #pragma unroll
    for (int v = 0; v < 8; ++v) {
        h1[(node0 + v + 8 * half) * F1 + l16] = c[v];
    }
}

// ---- layer-1 edge scatter: 16 consecutive lanes per edge ------------------
// agg1[col][f] += h1[row][f] * dinv[row]*dinv[col]
__global__ void gcn_scatter1_kernel(const int* __restrict__ rows,
                                    const int* __restrict__ cols,
                                    const float* __restrict__ h1,
                                    const float* __restrict__ dinv,
                                    float* __restrict__ agg1,
                                    int n_edges) {
    long long gid = (long long)blockIdx.x * blockDim.x + threadIdx.x;
    int e = (int)(gid >> 4);
    int f = (int)(gid & 15);
    if (e >= n_edges) return;
    int r = rows[e];
    int cidx = cols[e];
    float norm = dinv[r] * dinv[cidx];
    atomic_add_f32(&agg1[cidx * F1 + f], h1[r * F1 + f] * norm);
}

// ---- fused: add self-loop + b1, ReLU, then h2 = out1 @ W2 -----------------
__global__ void gcn_layer1_finish_kernel(const float* __restrict__ h1,
                                         const float* __restrict__ agg1,
                                         const float* __restrict__ dinv,
                                         const float* __restrict__ b1,
                                         const float* __restrict__ W2, // [16,2]
                                         float* __restrict__ h2,       // [N,2]
                                         int n) {
    int i = blockIdx.x * blockDim.x + threadIdx.x;
    if (i >= n) return;
    float di = dinv[i];
    float sw = di * di;                     // self-loop norm = dinv[i]^2
    float a0 = 0.0f, a1 = 0.0f;
    #pragma unroll
    for (int f = 0; f < F1; ++f) {
        float v = agg1[i * F1 + f] + h1[i * F1 + f] * sw + b1[f];
        v = v > 0.0f ? v : 0.0f;            // ReLU
        a0 = fmaf(v, W2[f * F2 + 0], a0);
        a1 = fmaf(v, W2[f * F2 + 1], a1);
    }
    h2[i * F2 + 0] = a0;
    h2[i * F2 + 1] = a1;
}

// ---- layer-2 edge scatter: 2 consecutive lanes per edge -------------------
__global__ void gcn_scatter2_kernel(const int* __restrict__ rows,
                                    const int* __restrict__ cols,
                                    const float* __restrict__ h2,
                                    const float* __restrict__ dinv,
                                    float* __restrict__ agg2,
                                    int n_edges) {
    long long gid = (long long)blockIdx.x * blockDim.x + threadIdx.x;
    int e = (int)(gid >> 1);
    int f = (int)(gid & 1);
    if (e >= n_edges) return;
    int r = rows[e];
    int cidx = cols[e];
    float norm = dinv[r] * dinv[cidx];
    atomic_add_f32(&agg2[cidx * F2 + f], h2[r * F2 + f] * norm);
}

// ---- final: self-loop + b2, log_softmax over 2 classes --------------------
__global__ void gcn_final_kernel(const float* __restrict__ h2,
                                 const float* __restrict__ agg2,
                                 const float* __restrict__ dinv,
                                 const float* __restrict__ b2,
                                 float* __restrict__ out,
                                 int n) {
    int i = blockIdx.x * blockDim.x + threadIdx.x;
    if (i >= n) return;
    float di = dinv[i];
    float sw = di * di;
    float a = agg2[i * F2 + 0] + h2[i * F2 + 0] * sw + b2[0];
    float b = agg2[i * F2 + 1] + h2[i * F2 + 1] * sw + b2[1];
    float m = fmaxf(a, b);
    float lse = m + logf(expf(a - m) + expf(b - m));
    out[i * F2 + 0] = a - lse;
    out[i * F2 + 1] = b - lse;
}

extern "C" void kernel_launch(void* const* d_in, const int* in_sizes, int n_in,
                              void* d_out, int out_size, void* d_ws, size_t ws_size,
                              hipStream_t stream) {
    const float* x   = (const float*)d_in[0];          // [N,25]
    const int*   ei  = (const int*)d_in[1];            // [2,E] int32
    const float* W1  = (const float*)d_in[2];          // [25,16]
    const float* b1  = (const float*)d_in[3];          // [16]
    const float* W2  = (const float*)d_in[4];          // [16,2]
    const float* b2  = (const float*)d_in[5];          // [2]
    float*       out = (float*)d_out;                  // [N,2]

    const int N = in_sizes[0] / FIN;                   // 100000
    const int E = in_sizes[1] / 2;                     // 3200000
    const int* rows = ei;                              // sources
    const int* cols = ei + E;                          // targets

    // Workspace layout (floats): 38*N total = 15.2 MB
    float* ws   = (float*)d_ws;
    float* deg  = ws;                 // [N]
    float* dinv = ws + (size_t)N;     // [N]
    float* h1   = ws + (size_t)2 * N; // [N,16]
    float* agg1 = ws + (size_t)18 * N;// [N,16]
    float* h2   = ws + (size_t)34 * N;// [N,2]
    float* agg2 = ws + (size_t)36 * N;// [N,2]

    const int BS = 256;  // 8 waves per workgroup on wave32

    // 1) init accumulators + self-loop degree
    gcn_init_kernel<<<(16 * N + BS - 1) / BS, BS, 0, stream>>>(deg, agg1, agg2, N);

    // 2) in-degree via L2 atomics
    gcn_degree_kernel<<<(E + BS - 1) / BS, BS, 0, stream>>>(cols, deg, E);

    // 3) dinv = rsqrt(deg)
    gcn_dinv_kernel<<<(N + BS - 1) / BS, BS, 0, stream>>>(deg, dinv, N);

    // 4) h1 = x @ W1 via fp32 WMMA (1 tile of 16 nodes per wave)
    {
        int tiles = (N + 15) / 16;                 // 6250
        int waves_per_block = BS / 32;             // 8
        int blocks = (tiles + waves_per_block - 1) / waves_per_block;
        gcn_xw1_wmma_kernel<<<blocks, BS, 0, stream>>>(x, W1, h1, N);
    }

    // 5) layer-1 aggregation: 16 lanes per edge (contiguous 64B gather/scatter)
    {
        long long tot = (long long)E * 16;
        int blocks = (int)((tot + BS - 1) / BS);
        gcn_scatter1_kernel<<<blocks, BS, 0, stream>>>(rows, cols, h1, dinv, agg1, E);
    }

    // 6) self-loop + bias + ReLU fused with h2 = out1 @ W2
    gcn_layer1_finish_kernel<<<(N + BS - 1) / BS, BS, 0, stream>>>(
        h1, agg1, dinv, b1, W2, h2, N);

    // 7) layer-2 aggregation: 2 lanes per edge
    {
        long long tot = (long long)E * 2;
        int blocks = (int)((tot + BS - 1) / BS);
        gcn_scatter2_kernel<<<blocks, BS, 0, stream>>>(rows, cols, h2, dinv, agg2, E);
    }

    // 8) self-loop + bias + log_softmax
    gcn_final_kernel<<<(N + BS - 1) / BS, BS, 0, stream>>>(h2, agg2, dinv, b2, out, N);
}